// Fixed_Mesh_2D_35871566856971
// MI455X (gfx1250) — compile-verified
//
#include <hip/hip_runtime.h>
#include <stdint.h>

// Fixed problem size (N_SIDE = 90)
#define NPTS   8100
#define NTRI   15842
#define M_ELEMS     65610000LL            // 8100*8100 floats
#define FILL_UNITS  16402500LL            // M_ELEMS / 4  (16-byte units)
#define TRI_OFF     65610000LL            // M | tri | slopes, concatenated
#define SLO_OFF     (65610000LL + 6LL * 15842LL)

typedef __attribute__((ext_vector_type(2))) float v2f;
typedef __attribute__((ext_vector_type(8))) float v8f;

// -------------------------------------------------------------------------
// Kernel 1: zero-fill the dense 262.4 MB matrix M using CDNA5 async stores
// (global_store_async_from_lds_b128): each lane streams its zeroed 16B LDS
// slot to memory; ops queue on ASYNCcnt (6-bit, HW throttles at 63
// outstanding), final s_wait_asynccnt 0 drains before endpgm.
// -------------------------------------------------------------------------
__global__ __launch_bounds__(256) void fem_fill_zero_async(float* __restrict__ M)
{
    __shared__ __align__(16) float zbuf[256 * 4];
    volatile float* zp = &zbuf[threadIdx.x * 4];
    zp[0] = 0.0f; zp[1] = 0.0f; zp[2] = 0.0f; zp[3] = 0.0f;
    // LDS aperture: low 32 bits of the flat address are the LDS byte offset.
    unsigned lds_addr = (unsigned)(uintptr_t)(&zbuf[threadIdx.x * 4]);
    asm volatile("s_wait_dscnt 0" ::: "memory");       // LDS zeros visible to async engine

    unsigned long long base = (unsigned long long)(uintptr_t)M;
    long long tid    = (long long)blockIdx.x * blockDim.x + threadIdx.x;
    long long stride = (long long)gridDim.x * blockDim.x;
    for (long long u = tid; u < FILL_UNITS; u += stride) {
        int voff = (int)(u << 4);                      // < 2^31, fits signed I32 vaddr
        asm volatile("global_store_async_from_lds_b128 %0, %1, %2"
                     :: "v"(voff), "v"(lds_addr), "s"(base) : "memory");
    }
    asm volatile("s_wait_asynccnt 0" ::: "memory");
}

// -------------------------------------------------------------------------
// Kernel 2: per-triangle geometry + WMMA Gram product + scatter assembly.
// 5 triangles per wave per v_wmma_f32_16x16x4_f32 (block-diagonal Gram).
// -------------------------------------------------------------------------
__global__ __launch_bounds__(256) void fem_tri_wmma(const float* __restrict__ pts,
                                                    const int*   __restrict__ map,
                                                    float* __restrict__ out)
{
    __shared__ float cbuf[8 * 256];                    // 8 waves * 16x16 f32 tile
    const int lane  = threadIdx.x & 31;
    const int wv    = threadIdx.x >> 5;
    const int wgid  = blockIdx.x * 8 + wv;
    const int tbase = wgid * 5;

    const bool active = (lane < 5) && (tbase + lane < NTRI);
    int t = tbase + lane; if (t >= NTRI) t = NTRI - 1;  // clamp: keep loads valid, no divergence

    const int i0 = map[3 * t + 0], i1 = map[3 * t + 1], i2 = map[3 * t + 2];
    const float x0 = pts[2 * i0], y0 = pts[2 * i0 + 1];
    const float x1 = pts[2 * i1], y1 = pts[2 * i1 + 1];
    const float x2 = pts[2 * i2], y2 = pts[2 * i2 + 1];

    // A = [[1,x0,y0],[1,x1,y1],[1,x2,y2]];  slopes = A^{-1} (adjugate / det)
    const float det  = (x1 - x0) * (y2 - y0) - (x2 - x0) * (y1 - y0);
    const float rdet = 1.0f / det;
    const float r00 = (x1 * y2 - y1 * x2) * rdet;      // row 0 of A^{-1}
    const float r01 = (y0 * x2 - x0 * y2) * rdet;
    const float r02 = (x0 * y1 - y0 * x1) * rdet;
    const float s00 = (y1 - y2) * rdet, s01 = (y2 - y0) * rdet, s02 = (y0 - y1) * rdet; // row 1
    const float s10 = (x2 - x1) * rdet, s11 = (x0 - x2) * rdet, s12 = (x1 - x0) * rdet; // row 2
    const float area = 0.5f * fabsf(det);

    if (active) {
        float* triO = out + TRI_OFF + 6LL * t;
        triO[0] = x0; triO[1] = y0; triO[2] = x1; triO[3] = y1; triO[4] = x2; triO[5] = y2;
        float* slo = out + SLO_OFF + 9LL * t;
        slo[0] = r00; slo[1] = r01; slo[2] = r02;
        slo[3] = s00; slo[4] = s01; slo[5] = s02;
        slo[6] = s10; slo[7] = s11; slo[8] = s12;
    }

    // Build the shared A/B operand for V_WMMA_F32_16X16X4_F32.
    // Lane m = 3*tl + a (m < 15) carries (s[0][a], s[1][a]) of local triangle tl.
    // Lanes 16..31 (K=2,3) and lane 15 are zero. A and B layouts coincide here.
    const int tl = lane / 3;
    const int a  = lane - 3 * tl;
    const float q00 = __shfl(s00, tl, 32), q01 = __shfl(s01, tl, 32), q02 = __shfl(s02, tl, 32);
    const float q10 = __shfl(s10, tl, 32), q11 = __shfl(s11, tl, 32), q12 = __shfl(s12, tl, 32);
    float A0 = (a == 0) ? q00 : ((a == 1) ? q01 : q02);
    float A1 = (a == 0) ? q10 : ((a == 1) ? q11 : q12);
    const bool rowvalid = (lane < 15) && (tbase + tl < NTRI);
    A0 = rowvalid ? A0 : 0.0f;
    A1 = rowvalid ? A1 : 0.0f;

    v2f ab; ab[0] = A0; ab[1] = A1;
    v8f c = {};
    // D = A x B + 0 : diagonal 3x3 blocks of D are the per-triangle Gram matrices
    c = __builtin_amdgcn_wmma_f32_16x16x4_f32(false, ab, false, ab,
                                              (short)0, c, false, false);

    // Spill C (documented 16x16 f32 C/D layout) to this wave's LDS tile.
    float* myc = &cbuf[wv * 256];
    const int rbase = (lane >= 16) ? 8 : 0;
    const int col   = lane & 15;
#pragma unroll
    for (int v = 0; v < 8; ++v) myc[(v + rbase) * 16 + col] = c[v];
    __builtin_amdgcn_wave_barrier();
    asm volatile("s_wait_dscnt 0" ::: "memory");       // wave-local LDS RAW

    if (active) {
        const int r = 3 * lane;                        // my diagonal block origin
        const float G00 = myc[(r + 0) * 16 + (r + 0)] * area;
        const float G11 = myc[(r + 1) * 16 + (r + 1)] * area;
        const float G22 = myc[(r + 2) * 16 + (r + 2)] * area;
        const float G01 = myc[(r + 0) * 16 + (r + 1)] * area;
        const float G12 = myc[(r + 1) * 16 + (r + 2)] * area;
        const float G20 = myc[(r + 2) * 16 + (r + 0)] * area;

        float* M = out;
        const long long nn = (long long)NPTS;
        // reference: M = -(scatter_add), symmetric assembly
        atomicAdd(&M[(long long)i0 * nn + i0], -G00);
        atomicAdd(&M[(long long)i1 * nn + i1], -G11);
        atomicAdd(&M[(long long)i2 * nn + i2], -G22);
        atomicAdd(&M[(long long)i0 * nn + i1], -G01);
        atomicAdd(&M[(long long)i1 * nn + i2], -G12);
        atomicAdd(&M[(long long)i2 * nn + i0], -G20);
        atomicAdd(&M[(long long)i1 * nn + i0], -G01);
        atomicAdd(&M[(long long)i2 * nn + i1], -G12);
        atomicAdd(&M[(long long)i0 * nn + i2], -G20);
    }
}

extern "C" void kernel_launch(void* const* d_in, const int* in_sizes, int n_in,
                              void* d_out, int out_size, void* d_ws, size_t ws_size,
                              hipStream_t stream)
{
    (void)in_sizes; (void)n_in; (void)out_size; (void)d_ws; (void)ws_size;
    const float* pts = (const float*)d_in[0];   // mesh_points (N,2) f32
    const int*   map = (const int*)d_in[1];     // cell_node_map (T,3) int
    float*       out = (float*)d_out;           // [M | tri | slopes] f32

    // 1) zero M (bandwidth-bound: ~262 MB, ~11 us at 23.3 TB/s)
    fem_fill_zero_async<<<2048, 256, 0, stream>>>(out);

    // 2) per-triangle compute + WMMA Gram + scatter (stream-ordered after fill)
    const int waves  = (NTRI + 4) / 5;          // 3169 waves of 5 triangles
    const int blocks = (waves + 7) / 8;         // 8 waves per 256-thread block
    fem_tri_wmma<<<blocks, 256, 0, stream>>>(pts, map, out);
}